// GraphAttentionConv_80599356277298
// MI455X (gfx1250) — compile-verified
//
#include <hip/hip_runtime.h>
#include <hip/hip_bf16.h>
#include <math.h>

#define EMB 64
#define NEG_SLOPE 0.2f

typedef __attribute__((ext_vector_type(2))) float v2f;
typedef __attribute__((ext_vector_type(8))) float v8f;

// Float atomic max via signed/unsigned integer atomics (IEEE-754 order trick).
// Works with m initialized to -inf (0xFF800000).
__device__ __forceinline__ void atomicMaxFloat(float* addr, float v) {
  if (v >= 0.0f) {
    atomicMax((int*)addr, __float_as_int(v));
  } else {
    atomicMin((unsigned int*)addr, __float_as_uint(v));
  }
}

__global__ void fill_kernel(float* __restrict__ p, float v, int n) {
  int i = blockIdx.x * blockDim.x + threadIdx.x;
  int stride = gridDim.x * blockDim.x;
  for (; i < n; i += stride) p[i] = v;
}

// ---------------------------------------------------------------------------
// Kernel 1: per-edge attention scores via f32 WMMA.
// One wave32 handles 16 edges: A = 16 src rows (16x64), B = 16 dst rows
// transposed (64x16); 16 chained V_WMMA_F32_16X16X4_F32 steps; the 16 scores
// are the diagonal of the 16x16 f32 accumulator.
//
// 32-bit A(16x4) / B(4x16) lane layout (ISA 7.12.2): lane L supplies
// row idx[L%16], K = 2*(L/16) + {vgpr0, vgpr1}  -> one b64 load per step.
// Diagonal (i,i): i<8 -> lane i, acc[i];  i>=8 -> lane i+8, acc[i-8].
// ---------------------------------------------------------------------------
__global__ void gat_score_wmma(const float* __restrict__ src_emb,
                               const float* __restrict__ dst_emb,
                               const int* __restrict__ src,
                               const int* __restrict__ dst,
                               float* __restrict__ e_out,
                               float* __restrict__ m_out,
                               int n_edges) {
  const int lane = threadIdx.x & 31;
  const int wave = (blockIdx.x * blockDim.x + threadIdx.x) >> 5;
  const int ebase = wave << 4;               // 16 edges per wave
  if (ebase >= n_edges) return;              // wave-uniform exit (EXEC stays full)

  const int slot_ld = lane & 15;             // which edge-slot this lane loads
  const int khalf   = lane >> 4;             // 0 or 1 (K pair within step)
  int ei = ebase + slot_ld;
  if (ei > n_edges - 1) ei = n_edges - 1;    // clamp (cndmask, no EXEC change)
  const long srow = (long)src[ei] * EMB;
  const long drow = (long)dst[ei] * EMB;

  v8f acc = {};
#pragma unroll
  for (int k = 0; k < 16; ++k) {
    const int col = 4 * k + 2 * khalf;
    v2f a = *(const v2f*)(src_emb + srow + col);
    v2f b = *(const v2f*)(dst_emb + drow + col);
    acc = __builtin_amdgcn_wmma_f32_16x16x4_f32(
        /*neg_a=*/false, a, /*neg_b=*/false, b,
        /*c_mod=*/(short)0, acc, /*reuse_a=*/false, /*reuse_b=*/false);
  }

  // Extract diagonal, leaky-ReLU, store score + running segment max.
  const bool has = (lane < 8) || (lane >= 16 && lane < 24);
  if (!has) return;
  const int r    = lane & 7;
  const int slot = (lane < 8) ? lane : (lane - 8);
  const int edge = ebase + slot;
  if (edge >= n_edges) return;

  float v = acc[0];
#pragma unroll
  for (int i = 1; i < 8; ++i)
    if (r == i) v = acc[i];

  v = (v >= 0.0f) ? v : NEG_SLOPE * v;       // leaky_relu
  e_out[edge] = v;
  atomicMaxFloat(&m_out[dst[edge]], v);
}

// ---------------------------------------------------------------------------
// Kernel 2: w = exp(e - m[dst]); s[dst] += w   (overwrites e with w in place)
// ---------------------------------------------------------------------------
__global__ void gat_weight(const int* __restrict__ dst,
                           const float* __restrict__ m,
                           float* __restrict__ e_w,
                           float* __restrict__ s,
                           int n_edges) {
  int i = blockIdx.x * blockDim.x + threadIdx.x;
  if (i >= n_edges) return;
  int d = dst[i];
  float w = expf(e_w[i] - m[d]);
  e_w[i] = w;
  unsafeAtomicAdd(&s[d], w);
}

// ---------------------------------------------------------------------------
// Kernel 3: out[dst] += (w/s[dst]) * src_row.  One wave32 per edge:
// 32 lanes x float2 = 64 cols; coalesced 256B row read, f32 atomic scatter.
// ---------------------------------------------------------------------------
__global__ void gat_aggregate(const float* __restrict__ src_emb,
                              const int* __restrict__ src,
                              const int* __restrict__ dst,
                              const float* __restrict__ w,
                              const float* __restrict__ s,
                              float* __restrict__ out,
                              int n_edges) {
  const int lane = threadIdx.x & 31;
  const int e = (blockIdx.x * blockDim.x + threadIdx.x) >> 5;
  if (e >= n_edges) return;
  const int d = dst[e];
  const float a = w[e] / s[d];
  const long srow = (long)src[e] * EMB;
  v2f x = *(const v2f*)(src_emb + srow + 2 * lane);
  float* o = out + (long)d * EMB + 2 * lane;
  unsafeAtomicAdd(o,     a * x.x);
  unsafeAtomicAdd(o + 1, a * x.y);
}

// ---------------------------------------------------------------------------

static inline int capped_blocks(long n, int tpb) {
  long b = (n + tpb - 1) / tpb;
  return (int)(b > 4096 ? 4096 : (b < 1 ? 1 : b));
}

extern "C" void kernel_launch(void* const* d_in, const int* in_sizes, int n_in,
                              void* d_out, int out_size, void* d_ws, size_t ws_size,
                              hipStream_t stream) {
  const float* user_emb = (const float*)d_in[0];
  const float* item_emb = (const float*)d_in[1];
  const int* ui_src = (const int*)d_in[2];
  const int* ui_dst = (const int*)d_in[3];
  const int* iu_src = (const int*)d_in[4];
  const int* iu_dst = (const int*)d_in[5];

  const int NU = in_sizes[0] / EMB;   // 100000
  const int NI = in_sizes[1] / EMB;   // 100000
  const int E1 = in_sizes[2];         // 1000000 (user->item)
  const int E2 = in_sizes[4];         // 1000000 (item->user)

  float* out_item = (float*)d_out;                       // [NI, 64]
  float* out_user = (float*)d_out + (size_t)NI * EMB;    // [NU, 64]

  // Workspace layout (floats): e/w per edge, then m and s per node.
  size_t need = ((size_t)E1 + E2 + 2 * ((size_t)NI + NU)) * sizeof(float);
  if (ws_size < need) return;
  float* ws     = (float*)d_ws;
  float* e1     = ws;                 // E1  (scores -> weights, ui)
  float* e2     = e1 + E1;            // E2  (scores -> weights, iu)
  float* m_item = e2 + E2;            // NI
  float* m_user = m_item + NI;        // NU   (contiguous with m_item)
  float* s_item = m_user + NU;        // NI
  float* s_user = s_item + NI;        // NU   (contiguous with s_item)

  const int TPB = 256;                // 8 wave32s per block

  // --- init: out = 0, m = -inf, s = 0 ---
  fill_kernel<<<capped_blocks(out_size, TPB), TPB, 0, stream>>>((float*)d_out, 0.0f, out_size);
  fill_kernel<<<capped_blocks(NI + NU, TPB), TPB, 0, stream>>>(m_item, -INFINITY, NI + NU);
  fill_kernel<<<capped_blocks(NI + NU, TPB), TPB, 0, stream>>>(s_item, 0.0f, NI + NU);

  // --- phase 1: WMMA scores + segment max ---
  {
    long waves1 = (E1 + 15) / 16, waves2 = (E2 + 15) / 16;
    int b1 = (int)((waves1 * 32 + TPB - 1) / TPB);
    int b2 = (int)((waves2 * 32 + TPB - 1) / TPB);
    gat_score_wmma<<<b1, TPB, 0, stream>>>(user_emb, item_emb, ui_src, ui_dst, e1, m_item, E1);
    gat_score_wmma<<<b2, TPB, 0, stream>>>(item_emb, user_emb, iu_src, iu_dst, e2, m_user, E2);
  }

  // --- phase 2: softmax numerators + denominators ---
  gat_weight<<<(E1 + TPB - 1) / TPB, TPB, 0, stream>>>(ui_dst, m_item, e1, s_item, E1);
  gat_weight<<<(E2 + TPB - 1) / TPB, TPB, 0, stream>>>(iu_dst, m_user, e2, s_user, E2);

  // --- phase 3: weighted scatter-accumulate (one wave per edge) ---
  {
    int b1 = (int)(((long)E1 * 32 + TPB - 1) / TPB);
    int b2 = (int)(((long)E2 * 32 + TPB - 1) / TPB);
    gat_aggregate<<<b1, TPB, 0, stream>>>(user_emb, ui_src, ui_dst, e1, s_item, out_item, E1);
    gat_aggregate<<<b2, TPB, 0, stream>>>(item_emb, iu_src, iu_dst, e2, s_user, out_user, E2);
  }
}